// HierarchicalReconstructionModule_51324859187633
// MI455X (gfx1250) — compile-verified
//
#include <hip/hip_runtime.h>

// HierarchicalReconstruction on gfx1250 (MI455X).
//
// Closed form: per bead n,
//   rel[j]  = normalize(node_output[n,j,:]) * bond_lengths[type[n], j]
//   P[k]    = sum_j T[k,j] * rel[j]          (T = anchor-chain reachability, 12x12)
//   out[k]  = (2 - sum_j w[j]) * bead_pos + P[k] - sum_j w[j]*P[j]
//
// P = T @ rel^T via chained V_WMMA_F32_16X16X4_F32 (K=12 -> 3 steps).
// One wave handles 5 beads: B columns = (bead, xyz) pairs (15 used, 1 pad
// column that computes harmless garbage and is never stored). A rows = T
// (rows 12..15 zero). T is bead-independent, derived by walking
// anchor_mask[bead0, level0, :].
//
// LDS rel layout is [bead][comp][j] so each lane's two K-values per WMMA
// chunk are adjacent -> one ds_load_b64 per chunk, unconditional (indices
// clamped, never predicated -> no exec-mask branches in the WMMA pipeline).

typedef __attribute__((ext_vector_type(2))) float v2f;
typedef __attribute__((ext_vector_type(8))) float v8f;

#define K_SLOTS          12
#define BEADS_PER_WAVE   5
#define WAVES_PER_BLOCK  4
#define BEADS_PER_BLOCK  (BEADS_PER_WAVE * WAVES_PER_BLOCK)  // 20

__global__ __launch_bounds__(128) void hier_recon_kernel(
    const float* __restrict__ node_output,   // (N, 36)
    const float* __restrict__ bead_pos,      // (N, 3)
    const float* __restrict__ weights,       // (N, 12)
    const float* __restrict__ bond_lengths,  // (33, 12, 1)
    const int*   __restrict__ bead_types,    // (N,)
    const int*   __restrict__ anchor_mask,   // (N, 4, 12); [0,0,k] = anchor_slot[k]
    float*       __restrict__ out,           // (N*12, 3)
    int n_beads)
{
    // rel staged as [n_loc][comp][j] : j contiguous, row = 48 B (16-aligned).
    __shared__ float s_rel[BEADS_PER_BLOCK * 3 * K_SLOTS];  // 720 floats

    const int t = threadIdx.x;
    const int blockBead = blockIdx.x * BEADS_PER_BLOCK;

    // ---- Stage 1: rel = normalize(node_output) * bond_length -> LDS ----
    for (int p = t; p < BEADS_PER_BLOCK * K_SLOTS; p += 128) {
        int n_loc = p / K_SLOTS;
        int j     = p % K_SLOTS;
        int n     = blockBead + n_loc;
        float rx = 0.f, ry = 0.f, rz = 0.f;
        if (n < n_beads) {
            const float* src = node_output + n * (K_SLOTS * 3) + j * 3;
            float x = src[0], y = src[1], z = src[2];
            float inv = 1.0f / (sqrtf(x * x + y * y + z * z) + 1e-5f);
            float bl  = bond_lengths[bead_types[n] * K_SLOTS + j];
            float s   = inv * bl;
            rx = x * s; ry = y * s; rz = z * s;
        }
        float* base = s_rel + n_loc * (3 * K_SLOTS) + j;
        base[0 * K_SLOTS] = rx;
        base[1 * K_SLOTS] = ry;
        base[2 * K_SLOTS] = rz;
    }
    __syncthreads();

    const int wave = t >> 5;
    const int lane = t & 31;
    const int hi   = lane >> 4;   // K-half of A/B, row-half of D
    const int lo   = lane & 15;   // A row m, B/D column

    // ---- A operand: row m of T as a bitmask (anchor chain walk, depth<=3) ----
    unsigned tmask = 0;
    if (lo < K_SLOTS) {
        int cur = lo;
        for (int it = 0; it < 8; ++it) {
            int a = anchor_mask[cur];        // bead 0, level 0 slice
            if (a == cur) break;             // root reached (anchor(0)==0)
            tmask |= 1u << cur;
            cur = a;
        }
    }

    // ---- B operand column -> (bead, xyz component); column 15 = pad ----
    const bool colActive = (lo < 15);
    const int  lb    = colActive ? (lo / 3) : 0;   // clamp: pad lane reads bead 0
    const int  comp  = colActive ? (lo % 3) : 0;
    const int  n_loc = wave * BEADS_PER_WAVE + lb;
    const int  n     = blockBead + n_loc;
    const bool beadOK = colActive && (n < n_beads);

    // Each chunk's two K-values are adjacent: relp[c*2 + hi] (ds_load_b64).
    const v2f* relp = (const v2f*)(s_rel + (n_loc * 3 + comp) * K_SLOTS);
    v2f b0 = relp[0 + hi];
    v2f b1 = relp[2 + hi];
    v2f b2 = relp[4 + hi];

    // A 16x4 f32 layout: lane L = row L%16, VGPR v = K (chunk*4 + v + 2*(L/16)).
    v2f a0, a1, a2;
    a0[0] = ((tmask >> (0 + 2 * hi)) & 1u)  ? 1.0f : 0.0f;
    a0[1] = ((tmask >> (1 + 2 * hi)) & 1u)  ? 1.0f : 0.0f;
    a1[0] = ((tmask >> (4 + 2 * hi)) & 1u)  ? 1.0f : 0.0f;
    a1[1] = ((tmask >> (5 + 2 * hi)) & 1u)  ? 1.0f : 0.0f;
    a2[0] = ((tmask >> (8 + 2 * hi)) & 1u)  ? 1.0f : 0.0f;
    a2[1] = ((tmask >> (9 + 2 * hi)) & 1u)  ? 1.0f : 0.0f;

    // ---- P = T @ rel^T : 3 chained V_WMMA_F32_16X16X4_F32 (K = 12) ----
    v8f acc = {};
    acc = __builtin_amdgcn_wmma_f32_16x16x4_f32(false, a0, false, b0,
                                                (short)0, acc, false, false);
    acc = __builtin_amdgcn_wmma_f32_16x16x4_f32(false, a1, false, b1,
                                                (short)0, acc, false, false);
    acc = __builtin_amdgcn_wmma_f32_16x16x4_f32(false, a2, false, b2,
                                                (short)0, acc, false, false);

    // ---- Center-of-mass shift: cmP = sum_m w[m]*P[m], sw = sum_m w[m] ----
    // D layout: VGPR r, lane L -> D[r + 8*(L/16), L%16].
    float partP = 0.f, partW = 0.f;
    if (beadOK) {
        // hi==0: rows 0..7 -> weights[n*12 + 0..7]  (two float4 loads)
        // hi==1: rows 8..11 -> weights[n*12 + 8..11] (one float4; quad 2 masked)
        const float* wbase = weights + n * K_SLOTS;
        float4 q0 = *(const float4*)(wbase + 8 * hi);   // rows r=0..3
        float4 q1 = *(const float4*)(wbase + 4);        // rows r=4..7 (hi==0 only)
        float hm  = hi ? 0.0f : 1.0f;
        partP = q0.x * acc[0] + q0.y * acc[1] + q0.z * acc[2] + q0.w * acc[3]
              + hm * (q1.x * acc[4] + q1.y * acc[5] + q1.z * acc[6] + q1.w * acc[7]);
        partW = q0.x + q0.y + q0.z + q0.w + hm * (q1.x + q1.y + q1.z + q1.w);
    }
    float cmP = partP + __shfl_xor(partP, 16, 32);
    float sw  = partW + __shfl_xor(partW, 16, 32);

    // ---- out[m] = (2 - sw)*bead_pos + P[m] - cmP ----
    if (beadOK) {
        float bp    = bead_pos[n * 3 + comp];
        float basev = (2.0f - sw) * bp - cmP;
        float* obase = out + n * (K_SLOTS * 3) + comp;
#pragma unroll
        for (int r = 0; r < 8; ++r) {
            int m2 = r + 8 * hi;
            if (m2 < K_SLOTS)
                obase[m2 * 3] = basev + acc[r];
        }
    }
}

extern "C" void kernel_launch(void* const* d_in, const int* in_sizes, int n_in,
                              void* d_out, int out_size, void* d_ws, size_t ws_size,
                              hipStream_t stream) {
    const float* node_output  = (const float*)d_in[0];
    const float* bead_pos     = (const float*)d_in[1];
    const float* weights      = (const float*)d_in[2];
    const float* bond_lengths = (const float*)d_in[3];
    const int*   bead_types   = (const int*)d_in[4];
    // d_in[5] b2a_idcs: identity mapping n*K+k (by construction) -> unused.
    // d_in[6] level_mask: redundant (level = anchor-chain depth) -> unused.
    const int*   anchor_mask  = (const int*)d_in[7];
    float* outp = (float*)d_out;

    const int n_beads = in_sizes[4];   // bead_types element count
    dim3 grid((n_beads + BEADS_PER_BLOCK - 1) / BEADS_PER_BLOCK);
    hier_recon_kernel<<<grid, 128, 0, stream>>>(
        node_output, bead_pos, weights, bond_lengths, bead_types,
        anchor_mask, outp, n_beads);
}